// HetConv_80281528696839
// MI455X (gfx1250) — compile-verified
//
#include <hip/hip_runtime.h>
#include <hip/hip_bf16.h>
#include <stdint.h>

// ---------------------------------------------------------------------------
// HetConv SpMM for MI455X (gfx1250):
//   out[row, off:off+128] += w * x[col, :]  per COO edge (E1=0.8M, E2=1.6M).
// Memory-bound: 0.61 GFLOP vs ~2.4GB of L2 traffic; x/out/edges all fit in
// the 192MB L2. The CDNA5 feature that matters is the Tensor Data Mover in
// GATHER mode: one tensor_load_to_lds pulls 16 indexed 512B rows of x into
// LDS (TENSORcnt), double-buffered against the fp32 L2-atomic scatter.
// ---------------------------------------------------------------------------

typedef unsigned int v4u __attribute__((ext_vector_type(4)));
typedef int          v4i __attribute__((ext_vector_type(4)));
typedef int          v8i __attribute__((ext_vector_type(8)));

#define D           128
#define OUT_STRIDE  256
#define TILE        16          // edges per TDM gather descriptor
#define WAVES_PB    4           // waves per block
#define CHUNK       64          // edges per wave (4 tiles)

__global__ void __launch_bounds__(256)
hetconv_zero_kernel(float4* __restrict__ out, int n4) {
    int i = blockIdx.x * blockDim.x + threadIdx.x;
    if (i < n4) out[i] = make_float4(0.f, 0.f, 0.f, 0.f);
}

struct Meta {
    int   r[TILE];
    int   c[TILE];
    float w[TILE];
};

// Wave-uniform metadata load for a tile of 16 edges (s_load path).
// Edges at position >= valid are padded: clamped index, weight 0.
__device__ __forceinline__ void load_meta(const int* __restrict__ row,
                                          const int* __restrict__ col,
                                          const float* __restrict__ w,
                                          int ec, int valid, int E, Meta& m) {
    if (ec + TILE <= E) {
#pragma unroll
        for (int q = 0; q < TILE / 4; ++q) {
            int4   rv = *(const int4*)(row + ec + 4 * q);
            int4   cv = *(const int4*)(col + ec + 4 * q);
            float4 wv = *(const float4*)(w + ec + 4 * q);
            m.r[4*q+0]=rv.x; m.r[4*q+1]=rv.y; m.r[4*q+2]=rv.z; m.r[4*q+3]=rv.w;
            m.c[4*q+0]=cv.x; m.c[4*q+1]=cv.y; m.c[4*q+2]=cv.z; m.c[4*q+3]=cv.w;
            m.w[4*q+0]=wv.x; m.w[4*q+1]=wv.y; m.w[4*q+2]=wv.z; m.w[4*q+3]=wv.w;
        }
    } else {
#pragma unroll
        for (int j = 0; j < TILE; ++j) {
            int e  = ec + j;
            int ee = e < E ? e : (E - 1);
            m.r[j] = row[ee]; m.c[j] = col[ee]; m.w[j] = w[ee];
        }
    }
#pragma unroll
    for (int j = 0; j < TILE; ++j)
        if (j >= valid) m.w[j] = 0.f;   // padded edges contribute +0.0
}

#if __has_builtin(__builtin_amdgcn_tensor_load_to_lds)
// Issue one TDM gather: 16 rows of x (indexed by m.c[0..15]) -> LDS.
__device__ __forceinline__ void tdm_gather16(const float* __restrict__ x,
                                             unsigned lds_addr,
                                             const Meta& m, int N) {
    unsigned long long ga = (unsigned long long)(const void*)x;

    // D# group 0 (ISA 8.3): gather_mode=1, 16-bit indices, count=1, type=2.
    v4u g0;
    g0.x = 0x80000001u;
    g0.y = lds_addr;                                   // LDS byte address
    g0.z = (unsigned)(ga & 0xFFFFFFFFu);               // global_addr[31:0]
    g0.w = (unsigned)((ga >> 32) & 0x1FFFFFFu) | (2u << 30);

    // D# group 1 (ISA 8.4): data_size=4B, dim0=128, dim1=N, tile0=128,
    // tile1=16 valid indices, dim0_stride=128.
    v8i g1;
    g1[0] = 0x00020000;                                // data_size=2 (4B)
    g1[1] = (int)(((unsigned)D & 0xFFFFu) << 16);      // tensor_dim0 lo16
    g1[2] = (int)(((unsigned)N & 0xFFFFu) << 16);      // dim0 hi=0 | dim1 lo16
    g1[3] = (int)((((unsigned)N >> 16) & 0xFFFFu) | ((unsigned)D << 16)); // dim1 hi | tile_dim0
    g1[4] = TILE;                                      // tile_dim1 = #indices
    g1[5] = D;                                         // tensor_dim0_stride lo32
    g1[6] = 0;
    g1[7] = 0;

    // D# groups 2/3 (ISA 8.7, gather mode): 16 packed 16-bit row indices.
    v4i g2, g3;
    g2.x = (m.c[1]  << 16) | (m.c[0]  & 0xFFFF);
    g2.y = (m.c[3]  << 16) | (m.c[2]  & 0xFFFF);
    g2.z = (m.c[5]  << 16) | (m.c[4]  & 0xFFFF);
    g2.w = (m.c[7]  << 16) | (m.c[6]  & 0xFFFF);
    g3.x = (m.c[9]  << 16) | (m.c[8]  & 0xFFFF);
    g3.y = (m.c[11] << 16) | (m.c[10] & 0xFFFF);
    g3.z = (m.c[13] << 16) | (m.c[12] & 0xFFFF);
    g3.w = (m.c[15] << 16) | (m.c[14] & 0xFFFF);

#if defined(__clang_major__) && __clang_major__ >= 23
    v8i zero = {};
    __builtin_amdgcn_tensor_load_to_lds(g0, g1, g2, g3, zero, 0);
#else
    __builtin_amdgcn_tensor_load_to_lds(g0, g1, g2, g3, 0);
#endif
}
#endif  // __has_builtin(tensor_load_to_lds)

__global__ void __launch_bounds__(WAVES_PB * 32)
hetconv_spmm_kernel(const float* __restrict__ x,
                    const int*   __restrict__ row,
                    const int*   __restrict__ col,
                    const float* __restrict__ w,
                    float*       __restrict__ out,
                    int E, int N, int outColOff) {
    const int lane  = threadIdx.x & 31;
    const int wv    = threadIdx.x >> 5;
    const int gwave = blockIdx.x * WAVES_PB + wv;

    const int start = gwave * CHUNK;
    if (start >= E) return;                    // wave-uniform exit
    const int count   = min(CHUNK, E - start);
    const int nTiles  = (count + TILE - 1) / TILE;

#if __has_builtin(__builtin_amdgcn_tensor_load_to_lds)
    // ---- TDM gather-mode pipeline: 16 rows per descriptor, 2 buffers ----
    __shared__ float sh[WAVES_PB][2][TILE][D];     // 4 * 2 * 8KB = 64KB
    const unsigned ldsBuf0 = (unsigned)(size_t)(&sh[wv][0][0][0]);
    const unsigned ldsBuf1 = (unsigned)(size_t)(&sh[wv][1][0][0]);

    Meta cur, nxt;
    {
        int ec = __builtin_amdgcn_readfirstlane(start);
        load_meta(row, col, w, ec, min(count, TILE), E, cur);
        tdm_gather16(x, ldsBuf0, cur, N);
    }

    int buf = 0;
    for (int t = 1; t < nTiles; ++t) {
        int ec    = __builtin_amdgcn_readfirstlane(start + t * TILE);
        int valid = min(count - t * TILE, TILE);
        load_meta(row, col, w, ec, valid, E, nxt);
        tdm_gather16(x, buf ? ldsBuf0 : ldsBuf1, nxt, N);

        // Oldest TDM op (current buffer) complete when TENSORcnt <= 1.
        asm volatile("s_wait_tensorcnt 0x1" ::: "memory");

#pragma unroll
        for (int j = 0; j < TILE; ++j) {
            const float4 v = *(const float4*)(&sh[wv][buf][j][lane * 4]);
            const float  wt = cur.w[j];
            float* o = out + (size_t)cur.r[j] * OUT_STRIDE + outColOff + lane * 4;
            atomicAdd(o + 0, v.x * wt);
            atomicAdd(o + 1, v.y * wt);
            atomicAdd(o + 2, v.z * wt);
            atomicAdd(o + 3, v.w * wt);
        }
#pragma unroll
        for (int j = 0; j < TILE; ++j) {
            cur.r[j] = nxt.r[j]; cur.c[j] = nxt.c[j]; cur.w[j] = nxt.w[j];
        }
        buf ^= 1;
    }

    asm volatile("s_wait_tensorcnt 0x0" ::: "memory");
#pragma unroll
    for (int j = 0; j < TILE; ++j) {
        const float4 v = *(const float4*)(&sh[wv][buf][j][lane * 4]);
        const float  wt = cur.w[j];
        float* o = out + (size_t)cur.r[j] * OUT_STRIDE + outColOff + lane * 4;
        atomicAdd(o + 0, v.x * wt);
        atomicAdd(o + 1, v.y * wt);
        atomicAdd(o + 2, v.z * wt);
        atomicAdd(o + 3, v.w * wt);
    }

#else
    // ---- Fallback: per-edge async VMEM->LDS gather (round-1 pipeline) ----
    __shared__ float sh[WAVES_PB][2][D];
    const unsigned lds0 = (unsigned)(size_t)(&sh[wv][0][lane * 4]);
    const unsigned lds1 = (unsigned)(size_t)(&sh[wv][1][lane * 4]);
    (void)nTiles; (void)N;

    int e  = start;
    int ec = __builtin_amdgcn_readfirstlane(e);
    int   r  = row[ec];
    int   c  = col[ec];
    float wt = w[ec];
    {
        unsigned long long ga =
            (unsigned long long)(const void*)(x + (size_t)c * D + lane * 4);
        asm volatile("global_load_async_to_lds_b128 %0, %1, off"
                     :: "v"(lds0), "v"(ga) : "memory");
    }
    int slot = 0;
    for (int en = e + 1; en < start + count; ++en) {
        int   ecn = __builtin_amdgcn_readfirstlane(en);
        int   rn  = row[ecn];
        int   cn  = col[ecn];
        float wn  = w[ecn];
        {
            unsigned long long gan =
                (unsigned long long)(const void*)(x + (size_t)cn * D + lane * 4);
            unsigned ldsn = slot ? lds0 : lds1;
            asm volatile("global_load_async_to_lds_b128 %0, %1, off"
                         :: "v"(ldsn), "v"(gan) : "memory");
        }
        asm volatile("s_wait_asynccnt 0x1" ::: "memory");
        {
            const float4 v = *(const float4*)(&sh[wv][slot][lane * 4]);
            float* o = out + (size_t)r * OUT_STRIDE + outColOff + lane * 4;
            atomicAdd(o + 0, v.x * wt);
            atomicAdd(o + 1, v.y * wt);
            atomicAdd(o + 2, v.z * wt);
            atomicAdd(o + 3, v.w * wt);
        }
        r = rn; wt = wn; slot ^= 1;
    }
    asm volatile("s_wait_asynccnt 0x0" ::: "memory");
    {
        const float4 v = *(const float4*)(&sh[wv][slot][lane * 4]);
        float* o = out + (size_t)r * OUT_STRIDE + outColOff + lane * 4;
        atomicAdd(o + 0, v.x * wt);
        atomicAdd(o + 1, v.y * wt);
        atomicAdd(o + 2, v.z * wt);
        atomicAdd(o + 3, v.w * wt);
    }
#endif
}

extern "C" void kernel_launch(void* const* d_in, const int* in_sizes, int n_in,
                              void* d_out, int out_size, void* d_ws, size_t ws_size,
                              hipStream_t stream) {
    const float* x    = (const float*)d_in[0];
    const int*   row1 = (const int*)  d_in[1];
    const int*   col1 = (const int*)  d_in[2];
    const float* w1   = (const float*)d_in[3];
    const int*   row2 = (const int*)  d_in[4];
    const int*   col2 = (const int*)  d_in[5];
    const float* w2   = (const float*)d_in[6];
    float*       out  = (float*)d_out;

    const int E1 = in_sizes[1];
    const int E2 = in_sizes[4];
    const int N  = in_sizes[0] / D;   // 50000

    // 1) zero the (poisoned) output: B128 stores.
    int n4 = out_size / 4;
    hetconv_zero_kernel<<<(n4 + 255) / 256, 256, 0, stream>>>((float4*)out, n4);

    // 2) scatter-add both edge lists (disjoint output column ranges).
    auto blocks_for = [](int E) {
        int waves = (E + CHUNK - 1) / CHUNK;
        return (waves + WAVES_PB - 1) / WAVES_PB;
    };
    hetconv_spmm_kernel<<<blocks_for(E1), WAVES_PB * 32, 0, stream>>>(
        x, row1, col1, w1, out, E1, N, /*outColOff=*/0);
    hetconv_spmm_kernel<<<blocks_for(E2), WAVES_PB * 32, 0, stream>>>(
        x, row2, col2, w2, out, E2, N, /*outColOff=*/D);
}